// Coord2HeatmapNet_88940182766251
// MI455X (gfx1250) — compile-verified
//
#include <hip/hip_runtime.h>
#include <hip/hip_bf16.h>

// Coord2Heatmap: out[b,c,j,i] = 10 * exp(-((x_i-cx)^2 + (y_j-cy)^2) / 2^-11)
// Separable Gaussian -> per-(b,c) rank-1 outer product fy (x) fx,
// computed on the matrix engine via V_WMMA_F32_16X16X4_F32 (K=0 only).
//
// B=64, C=68, S=128. Output 285 MB fp32 -> store-bandwidth bound (~12 us at
// 23.3 TB/s). NT stores to avoid thrashing the 192 MB L2 (no reuse).
// All LDS reads are unconditional (every lane has a valid address); lane
// masking is done with cndmask so no exec-masked DS blocks appear, and all
// fx segments are prefetched in one DS clause before the WMMA/store loop.

#define SS 128
#define NCLS 68
#define NB 64

typedef __attribute__((ext_vector_type(2))) float v2f;
typedef __attribute__((ext_vector_type(8))) float v8f;

__global__ __launch_bounds__(256) void coord2heatmap_wmma(
    const float* __restrict__ coords,  // (B, NCLS*2) flat; coords[2*bc]=cx, [2*bc+1]=cy
    float* __restrict__ out)           // (B, NCLS, S, S) flat
{
    __shared__ float fac[2 * SS];  // [0:128) = fx, [128:256) = 10*fy

    const int bc = blockIdx.x;          // b*NCLS + c
    const int t  = threadIdx.x;         // 0..255

    const float cx = coords[2 * bc + 0];
    const float cy = coords[2 * bc + 1];

    // ---- Phase 1: separable 1-D Gaussian factors into LDS ----
    // exp(-d^2 / 2^-11) = exp2(-d^2 * 2048 * log2(e))
    {
        const int   i   = t & (SS - 1);
        const float lin = ((float)i + 0.5f) * (1.0f / (float)SS);
        const float ctr = (t < SS) ? cx : cy;
        const float d   = lin - ctr;
        float f = __builtin_amdgcn_exp2f(d * d * -2954.6394436593f); // -2048*log2(e)
        if (t >= SS) f *= 10.0f;   // fold the *10 into the row factor
        fac[t] = f;
    }
    __syncthreads();

    // ---- Phase 2: 16x16 tiles via WMMA outer product ----
    const int  wave  = t >> 5;          // 0..7 -> row tile tj
    const int  lane  = t & 31;
    const bool lowhf = (lane >> 4) == 0;  // lanes 0-15 carry K=0 data
    const int  l16   = lane & 15;

    // Unconditional LDS reads (addresses valid for every lane), then cndmask.
    const float fyv = fac[SS + wave * 16 + l16];
    float fxv[8];
    #pragma unroll
    for (int ti = 0; ti < 8; ++ti)
        fxv[ti] = fac[ti * 16 + l16];

    // A-matrix 16x4 f32 (ISA layout): lanes 0-15 hold M=lane; VGPR0=K0, VGPR1=K1;
    // lanes 16-31 hold K2/K3. Only K=0 is populated with fy (pre-scaled by 10).
    v2f a;
    a.x = lowhf ? fyv : 0.0f;
    a.y = 0.0f;

    float* outbase = out + (size_t)bc * (SS * SS);
    const int rowoff = wave * 16 + (lowhf ? 0 : 8);  // D layout: lanes16-31 -> M=r+8

    #pragma unroll
    for (int ti = 0; ti < 8; ++ti) {
        // B-matrix 4x16 f32: only K=0 row populated with fx (VGPR0, lanes 0-15).
        v2f bm;
        bm.x = lowhf ? fxv[ti] : 0.0f;
        bm.y = 0.0f;

        v8f c = {0.f, 0.f, 0.f, 0.f, 0.f, 0.f, 0.f, 0.f};
        // D[M][N] = fy[M] * fx[N]  (rank-1 GEMM, K=0 column/row only)
        c = __builtin_amdgcn_wmma_f32_16x16x4_f32(
                /*neg_a=*/false, a, /*neg_b=*/false, bm,
                /*c_mod=*/(short)0, c, /*reuse_a=*/false, /*reuse_b=*/false);

        // f32 C/D layout: VGPR r -> row (rowoff + r), col = ti*16 + l16.
        float* p = outbase + (size_t)rowoff * SS + ti * 16 + l16;
        #pragma unroll
        for (int r = 0; r < 8; ++r)
            __builtin_nontemporal_store(c[r], p + (size_t)r * SS);
    }
}

extern "C" void kernel_launch(void* const* d_in, const int* in_sizes, int n_in,
                              void* d_out, int out_size, void* d_ws, size_t ws_size,
                              hipStream_t stream)
{
    const float* coords = (const float*)d_in[0];
    float* out = (float*)d_out;
    (void)in_sizes; (void)n_in; (void)d_ws; (void)ws_size; (void)out_size;

    coord2heatmap_wmma<<<dim3(NB * NCLS), dim3(256), 0, stream>>>(coords, out);
}